// Model_34772055229099
// MI455X (gfx1250) — compile-verified
//
#include <hip/hip_runtime.h>

typedef float v2f __attribute__((ext_vector_type(2)));
typedef float v8f __attribute__((ext_vector_type(8)));

#define NB 56      // (4 batches) * (14 channels) rows
#define DSTATE 8
#define DCONV 4

__device__ __forceinline__ float siluf(float x) { return x / (1.f + expf(-x)); }
__device__ __forceinline__ float gelu_tanh(float x) {
  const float k0 = 0.7978845608028654f, k1 = 0.044715f;
  return 0.5f * x * (1.f + tanhf(k0 * (x + k1 * x * x * x)));
}
__device__ __forceinline__ float softplusf(float x) {
  return (x > 20.f) ? x : log1pf(expf(x));
}
__device__ __forceinline__ float actf(float x, int act) {
  if (act == 1) return siluf(x);
  if (act == 2) return gelu_tanh(x);
  if (act == 3) return softplusf(x);
  return x;
}

// Branch-free guarded v2f load: clamp address in-bounds, zero via select.
__device__ __forceinline__ v2f ld2_guard(const float* __restrict__ p, size_t rowbase,
                                         bool rowok, int ka, int K) {
  int k0 = min(ka, K - 1);
  int k1 = min(ka + 1, K - 1);
  float x0 = p[rowbase + k0];
  float x1 = p[rowbase + k1];
  v2f r;
  r.x = (rowok && ka < K) ? x0 : 0.f;
  r.y = (rowok && ka + 1 < K) ? x1 : 0.f;
  return r;
}

// -----------------------------------------------------------------------------
// fp32 WMMA GEMM: Y[m,n] = act( (Cin? Cin[m,n]:0) + sum_k A[m,k]*W[n,k] + bias[n] )
// A: R x K (lda, aoff). W: N x K row-major (=> A @ W^T).
// 128-thread block = 4 waves in a 2x2 arrangement; each wave owns a 32x32 tile
// (2x2 of 16x16 WMMA subtiles, 4 independent accumulators -> no D->C RAW
// hazard, 4 v_wmma per K-step of 4 fed by just four b64 loads).
// All call-site R are multiples of 64; only N-edges / odd strides / K%4!=0 take
// the guarded loop, which is branch-free (clamped loads + cndmask selects).
// Fragment layouts per ISA 7.12.2:
//   A 16x4 : lane m=lane&15 holds K = 2*(lane>>4), +1   (one v2f load)
//   B 4x16 : mirrored, N across lanes                    (one v2f load)
//   C/D    : vgpr v -> M = v + 8*(lane>>4), N = lane&15
// -----------------------------------------------------------------------------
__global__ void __launch_bounds__(128)
gemm_wmma(const float* __restrict__ A, int lda, int aoff,
          const float* __restrict__ W, const float* __restrict__ bias,
          const float* __restrict__ Cin,
          float* __restrict__ Cout, int ldc, int coff,
          int R, int K, int N, int act)
{
  const int wave = threadIdx.x >> 5;
  const int lane = threadIdx.x & 31;
  const int mr = lane & 15;
  const int kh = lane >> 4;
  const int m0 = blockIdx.x * 64 + (wave & 1) * 32;
  const int n0 = blockIdx.y * 64 + (wave >> 1) * 32;
  if (m0 >= R || n0 >= N) return;            // wave-uniform

  const int row0 = m0 + mr, row1 = row0 + 16;
  const int nn0 = n0 + mr,  nn1 = nn0 + 16;

  const bool fast = (m0 + 32 <= R) && (n0 + 32 <= N) && ((K & 3) == 0) &&
                    ((lda & 1) == 0) && ((aoff & 1) == 0);

  v8f acc00 = {0.f,0.f,0.f,0.f,0.f,0.f,0.f,0.f};
  v8f acc01 = acc00, acc10 = acc00, acc11 = acc00;

  if (Cin) {                                  // init accumulators from Cin
#pragma unroll
    for (int v = 0; v < 8; ++v) {
      int ma = m0 + v + (kh << 3), mb = ma + 16;
      int ca = min(ma, R - 1), cb = min(mb, R - 1);
      int na = min(nn0, N - 1), nbn = min(nn1, N - 1);
      float v00 = Cin[(size_t)ca * ldc + coff + na];
      float v01 = Cin[(size_t)ca * ldc + coff + nbn];
      float v10 = Cin[(size_t)cb * ldc + coff + na];
      float v11 = Cin[(size_t)cb * ldc + coff + nbn];
      acc00[v] = (ma < R && nn0 < N) ? v00 : 0.f;
      acc01[v] = (ma < R && nn1 < N) ? v01 : 0.f;
      acc10[v] = (mb < R && nn0 < N) ? v10 : 0.f;
      acc11[v] = (mb < R && nn1 < N) ? v11 : 0.f;
    }
  }

  if (fast) {
    const float* Ap0 = A + (size_t)row0 * lda + aoff + (kh << 1);
    const float* Ap1 = Ap0 + (size_t)16 * lda;
    const float* Wp0 = W + (size_t)nn0 * K + (kh << 1);
    const float* Wp1 = Wp0 + (size_t)16 * K;
#pragma unroll 2
    for (int k = 0; k < K; k += 4) {
      v2f a0 = *(const v2f*)Ap0;
      v2f a1 = *(const v2f*)Ap1;
      v2f b0 = *(const v2f*)Wp0;
      v2f b1 = *(const v2f*)Wp1;
      acc00 = __builtin_amdgcn_wmma_f32_16x16x4_f32(false, a0, false, b0, (short)0,
                                                    acc00, false, false);
      acc01 = __builtin_amdgcn_wmma_f32_16x16x4_f32(false, a0, false, b1, (short)0,
                                                    acc01, false, false);
      acc10 = __builtin_amdgcn_wmma_f32_16x16x4_f32(false, a1, false, b0, (short)0,
                                                    acc10, false, false);
      acc11 = __builtin_amdgcn_wmma_f32_16x16x4_f32(false, a1, false, b1, (short)0,
                                                    acc11, false, false);
      Ap0 += 4; Ap1 += 4; Wp0 += 4; Wp1 += 4;
    }
  } else {
    const size_t ab0 = (size_t)min(row0, R - 1) * lda + aoff;
    const size_t ab1 = (size_t)min(row1, R - 1) * lda + aoff;
    const size_t wb0 = (size_t)min(nn0, N - 1) * K;
    const size_t wb1 = (size_t)min(nn1, N - 1) * K;
    const bool r0ok = row0 < R, r1ok = row1 < R, n0ok = nn0 < N, n1ok = nn1 < N;
    for (int k = 0; k < K; k += 4) {
      int ka = k + (kh << 1);
      v2f a0 = ld2_guard(A, ab0, r0ok, ka, K);
      v2f a1 = ld2_guard(A, ab1, r1ok, ka, K);
      v2f b0 = ld2_guard(W, wb0, n0ok, ka, K);
      v2f b1 = ld2_guard(W, wb1, n1ok, ka, K);
      acc00 = __builtin_amdgcn_wmma_f32_16x16x4_f32(false, a0, false, b0, (short)0,
                                                    acc00, false, false);
      acc01 = __builtin_amdgcn_wmma_f32_16x16x4_f32(false, a0, false, b1, (short)0,
                                                    acc01, false, false);
      acc10 = __builtin_amdgcn_wmma_f32_16x16x4_f32(false, a1, false, b0, (short)0,
                                                    acc10, false, false);
      acc11 = __builtin_amdgcn_wmma_f32_16x16x4_f32(false, a1, false, b1, (short)0,
                                                    acc11, false, false);
    }
  }

  // ---- epilogue: bias + activation + (guarded) stores, 2x2 subtiles ----
  float bv0 = 0.f, bv1 = 0.f;
  if (bias) {
    bv0 = bias[min(nn0, N - 1)];
    bv1 = bias[min(nn1, N - 1)];
  }
  if (fast) {
#pragma unroll
    for (int v = 0; v < 8; ++v) {
      size_t ra = (size_t)(m0 + v + (kh << 3)) * ldc + coff;
      size_t rb = ra + (size_t)16 * ldc;
      Cout[ra + nn0] = actf(acc00[v] + bv0, act);
      Cout[ra + nn1] = actf(acc01[v] + bv1, act);
      Cout[rb + nn0] = actf(acc10[v] + bv0, act);
      Cout[rb + nn1] = actf(acc11[v] + bv1, act);
    }
  } else {
#pragma unroll
    for (int v = 0; v < 8; ++v) {
      int ma = m0 + v + (kh << 3), mb = ma + 16;
      if (ma < R && nn0 < N) Cout[(size_t)ma * ldc + coff + nn0] = actf(acc00[v] + bv0, act);
      if (ma < R && nn1 < N) Cout[(size_t)ma * ldc + coff + nn1] = actf(acc01[v] + bv1, act);
      if (mb < R && nn0 < N) Cout[(size_t)mb * ldc + coff + nn0] = actf(acc10[v] + bv0, act);
      if (mb < R && nn1 < N) Cout[(size_t)mb * ldc + coff + nn1] = actf(acc11[v] + bv1, act);
    }
  }
}

// Depthwise causal (rev: anti-causal) conv over time, + bias, + SiLU.
__global__ void conv_silu(const float* __restrict__ xz, int ld, int off,
                          const float* __restrict__ w, const float* __restrict__ bias,
                          float* __restrict__ xc, int T, int di, int rev)
{
  int idx = blockIdx.x * blockDim.x + threadIdx.x;
  int total = NB * T * di;
  if (idx >= total) return;
  int c = idx % di;
  int t = (idx / di) % T;
  int br = idx / (di * T);
  const float* base = xz + (size_t)br * T * ld + off + c;
  float s = bias[c];
#pragma unroll
  for (int j = 0; j < DCONV; ++j) {
    int tt = rev ? (t + (DCONV - 1) - j) : (t - (DCONV - 1) + j);
    float v = base[(size_t)min(max(tt, 0), T - 1) * ld];
    s += (tt >= 0 && tt < T) ? w[c * DCONV + j] * v : 0.f;
  }
  xc[(size_t)idx] = siluf(s);
}

// Selective scan: one thread per (batch-row, channel). h[8] lives in registers.
__global__ void ssm_scan(const float* __restrict__ delta,
                         const float* __restrict__ xdb, int ldx, int dtr,
                         const float* __restrict__ xc,
                         const float* __restrict__ xz, int ldz, int zoff,
                         const float* __restrict__ A_log, const float* __restrict__ Dp,
                         float* __restrict__ y, int T, int di, int rev)
{
  int idx = blockIdx.x * blockDim.x + threadIdx.x;
  if (idx >= NB * di) return;
  int c = idx % di;
  int br = idx / di;
  float A[DSTATE], h[DSTATE];
#pragma unroll
  for (int s = 0; s < DSTATE; ++s) { A[s] = -expf(A_log[c * DSTATE + s]); h[s] = 0.f; }
  float Dv = Dp[c];
  for (int it = 0; it < T; ++it) {
    int t = rev ? (T - 1 - it) : it;
    size_t r = (size_t)br * T + t;
    float dv = delta[r * di + c];
    float xv = xc[r * di + c];
    const float* bm = xdb + r * ldx + dtr;
    float yv = 0.f;
#pragma unroll
    for (int s = 0; s < DSTATE; ++s) {
      h[s] = expf(dv * A[s]) * h[s] + dv * bm[s] * xv;
      yv += h[s] * bm[DSTATE + s];
    }
    yv += xv * Dv;
    float zv = xz[r * ldz + zoff + c];
    y[r * di + c] = yv * siluf(zv);
  }
}

// (NB,T,D) -> (NB,D,T)
__global__ void transpose_bt(const float* __restrict__ in, float* __restrict__ out,
                             int T, int D)
{
  int idx = blockIdx.x * blockDim.x + threadIdx.x;
  int total = NB * T * D;
  if (idx >= total) return;
  int c = idx % D;
  int t = (idx / D) % T;
  int br = idx / (D * T);
  out[((size_t)br * D + c) * T + t] = in[idx];
}

// LayerNorm over last dim (D in {32,64,128}), one wave per row.
// Optionally adds addA (same layout) and sT (transposed (NB,D,T) layout) first.
__global__ void __launch_bounds__(32)
ln_fuse(const float* __restrict__ xin, const float* __restrict__ addA,
        const float* __restrict__ sT, const float* __restrict__ g,
        const float* __restrict__ bta, float* __restrict__ xout, int T, int D)
{
  int r = blockIdx.x;                 // row in [0, NB*T)
  int br = r / T, t = r % T;
  int lane = threadIdx.x;
  int nper = D >> 5;                  // <= 4
  float vals[4];
  float sum = 0.f, sq = 0.f;
  for (int j = 0; j < nper; ++j) {
    int c = lane + (j << 5);
    float v = xin[(size_t)r * D + c];
    if (addA) v += addA[(size_t)r * D + c];
    if (sT)   v += sT[((size_t)br * D + c) * T + t];
    vals[j] = v;
    sum += v;
    sq += v * v;
  }
#pragma unroll
  for (int o = 16; o > 0; o >>= 1) {
    sum += __shfl_xor(sum, o, 32);
    sq  += __shfl_xor(sq,  o, 32);
  }
  float mean = sum / D;
  float var = sq / D - mean * mean;
  float inv = rsqrtf(var + 1e-5f);
  for (int j = 0; j < nper; ++j) {
    int c = lane + (j << 5);
    xout[(size_t)r * D + c] = (vals[j] - mean) * inv * g[c] + bta[c];
  }
}

// Patch embedding: x[bc,t,:] = patch_w @ x_enc[b, 16t:16t+16, ch] + patch_b + pos[t]
__global__ void patch_embed(const float* __restrict__ xe, const float* __restrict__ pw,
                            const float* __restrict__ pb, const float* __restrict__ pos,
                            float* __restrict__ x)
{
  int idx = blockIdx.x * blockDim.x + threadIdx.x;
  if (idx >= NB * 32 * 128) return;
  int dout = idx % 128;
  int t = (idx / 128) % 32;
  int bc = idx / (128 * 32);
  int b = bc / 14, ch = bc % 14;
  const float* src = xe + ((size_t)b * 512 + t * 16) * 14 + ch;
  float s = pb[dout] + pos[t * 128 + dout];
#pragma unroll
  for (int k = 0; k < 16; ++k) s += pw[dout * 16 + k] * src[(size_t)k * 14];
  x[idx] = s;
}

// Per-scale head: outs[r, sidx] = dot(x[r,:], w) + b
__global__ void __launch_bounds__(32)
head_dot(const float* __restrict__ x, const float* __restrict__ w,
         const float* __restrict__ b, float* __restrict__ outs, int K, int sidx)
{
  int r = blockIdx.x;
  float s = 0.f;
  for (int k = threadIdx.x; k < K; k += 32) s += x[(size_t)r * K + k] * w[k];
#pragma unroll
  for (int o = 16; o > 0; o >>= 1) s += __shfl_xor(s, o, 32);
  if (threadIdx.x == 0) outs[r * 3 + sidx] = s + b[0];
}

__global__ void final_proj(const float* __restrict__ outs, const float* __restrict__ fw,
                           const float* __restrict__ fb, float* __restrict__ out)
{
  int r = blockIdx.x * blockDim.x + threadIdx.x;
  if (r < NB)
    out[r] = outs[r * 3] * fw[0] + outs[r * 3 + 1] * fw[1] + outs[r * 3 + 2] * fw[2] + fb[0];
}

// -----------------------------------------------------------------------------
// Host orchestration
// -----------------------------------------------------------------------------
struct MambaP {
  const float *in_w, *conv_w, *conv_b, *x_w, *dt_w, *dt_b, *A_log, *Dp, *out_w;
};
struct LayerP {
  MambaP mt, ms;
  const float *ln1g, *ln1b, *ln2g, *ln2b, *w1, *b1, *w2, *b2;
};

static void launch_gemm(hipStream_t st, const float* A, int lda, int aoff,
                        const float* W, const float* bias, const float* Cin,
                        float* Cout, int ldc, int coff, int R, int K, int N, int act)
{
  dim3 grid((R + 63) / 64, (N + 63) / 64);
  gemm_wmma<<<grid, 128, 0, st>>>(A, lda, aoff, W, bias, Cin, Cout, ldc, coff,
                                  R, K, N, act);
}

// bimamba: out = mamba(x) + flip(mamba(flip(x)))   for x of shape (NB, T, Dm)
static void run_bimamba(hipStream_t st, const float* x, float* out, const MambaP& p,
                        int T, int Dm, float* XZ, float* XC, float* XDB,
                        float* DELTA, float* Y)
{
  int di = 4 * Dm;
  int dtr = (Dm + 15) / 16; if (dtr < 1) dtr = 1;
  int R = NB * T;
  int ldx = dtr + 2 * DSTATE;
  // shared in-projection (per-row op commutes with flip)
  launch_gemm(st, x, Dm, 0, p.in_w, nullptr, nullptr, XZ, 2 * di, 0, R, Dm, 2 * di, 0);
  for (int rev = 0; rev < 2; ++rev) {
    int tot = R * di;
    conv_silu<<<(tot + 255) / 256, 256, 0, st>>>(XZ, 2 * di, 0, p.conv_w, p.conv_b,
                                                 XC, T, di, rev);
    launch_gemm(st, XC, di, 0, p.x_w, nullptr, nullptr, XDB, ldx, 0, R, di, ldx, 0);
    launch_gemm(st, XDB, ldx, 0, p.dt_w, p.dt_b, nullptr, DELTA, di, 0, R, dtr, di, 3);
    int sc = NB * di;
    ssm_scan<<<(sc + 127) / 128, 128, 0, st>>>(DELTA, XDB, ldx, dtr, XC, XZ, 2 * di, di,
                                               p.A_log, p.Dp, Y, T, di, rev);
    // accumulate fwd + rev directly through the WMMA C operand
    launch_gemm(st, Y, di, 0, p.out_w, nullptr, rev ? out : nullptr, out, Dm, 0,
                R, di, Dm, 0);
  }
}

extern "C" void kernel_launch(void* const* d_in, const int* in_sizes, int n_in,
                              void* d_out, int out_size, void* d_ws, size_t ws_size,
                              hipStream_t stream)
{
  (void)in_sizes; (void)n_in; (void)out_size; (void)ws_size;

  // ---- walk flattened params in setup_inputs() dict order ----
  int idx = 0;
  auto nf = [&]() -> const float* { return (const float*)d_in[idx++]; };
  const float* x_enc   = nf();
  const float* patch_w = nf();
  const float* patch_b = nf();
  const float* posb    = nf();
  LayerP layers[3][4];
  const float *ng[3], *nbb[3], *hw[3], *hb[3], *mw[2], *mb[2];
  auto rm = [&](MambaP& m) {
    m.in_w = nf(); m.conv_w = nf(); m.conv_b = nf(); m.x_w = nf(); m.dt_w = nf();
    m.dt_b = nf(); m.A_log = nf(); m.Dp = nf(); m.out_w = nf();
  };
  for (int i = 0; i < 3; ++i) {
    for (int l = 0; l < 4; ++l) {
      LayerP& L = layers[i][l];
      rm(L.mt); rm(L.ms);
      L.ln1g = nf(); L.ln1b = nf(); L.ln2g = nf(); L.ln2b = nf();
      L.w1 = nf(); L.b1 = nf(); L.w2 = nf(); L.b2 = nf();
    }
    ng[i] = nf(); nbb[i] = nf();
  }
  for (int i = 0; i < 3; ++i) { hw[i] = nf(); hb[i] = nf(); }
  for (int i = 0; i < 2; ++i) { mw[i] = nf(); mb[i] = nf(); }
  const float* fw = nf();
  const float* fb = nf();

  // ---- workspace carve-up (sized for scale 0; later scales shrink) ----
  float* wsp = (float*)d_ws;
  size_t off = 0;
  auto alloc = [&](size_t n) { float* p = wsp + off; off += n; return p; };
  float* XA    = alloc(229376);   // x        (56,32,128)
  float* XB    = alloc(229376);   // merge ping-pong
  float* TB    = alloc(229376);   // bimamba-time output
  float* XT    = alloc(229376);   // x transposed (56,D,T)
  float* ST    = alloc(229376);   // bimamba-space output (transposed layout)
  float* TMP   = alloc(229376);   // FFN residual
  float* XZ    = alloc(1835008);  // in-proj out (max over mt/ms at scale 0)
  float* XC    = alloc(917504);   // conv+silu out
  float* XDB   = alloc(172032);   // x_proj out (dt|B|C)
  float* DELTA = alloc(917504);
  float* Y     = alloc(917504);
  float* H     = alloc(458752);   // FFN hidden (56*32,256)
  float* OUTS  = alloc(256);      // per-scale heads (56,3)

  // ---- forward ----
  patch_embed<<<(NB * 32 * 128 + 255) / 256, 256, 0, stream>>>(x_enc, patch_w,
                                                               patch_b, posb, XA);
  float* X = XA;
  float* Xalt = XB;
  int T = 32, D = 128;
  for (int i = 0; i < 3; ++i) {
    for (int l = 0; l < 4; ++l) {
      const LayerP& L = layers[i][l];
      // time-mixing bimamba on (56,T,D)
      run_bimamba(stream, X, TB, L.mt, T, D, XZ, XC, XDB, DELTA, Y);
      // space-mixing bimamba on (56,D,T)
      int tot = NB * T * D;
      transpose_bt<<<(tot + 255) / 256, 256, 0, stream>>>(X, XT, T, D);
      run_bimamba(stream, XT, ST, L.ms, D, T, XZ, XC, XDB, DELTA, Y);
      // x = LN(x + t + s^T)
      ln_fuse<<<NB * T, 32, 0, stream>>>(X, TB, ST, L.ln1g, L.ln1b, X, T, D);
      // FFN: h = gelu(x@w1^T+b1); x = LN(x + h@w2^T + b2)
      launch_gemm(stream, X, D, 0, L.w1, L.b1, nullptr, H, 256, 0, NB * T, D, 256, 2);
      launch_gemm(stream, H, 256, 0, L.w2, L.b2, X, TMP, D, 0, NB * T, 256, D, 0);
      ln_fuse<<<NB * T, 32, 0, stream>>>(TMP, nullptr, nullptr, L.ln2g, L.ln2b, X, T, D);
    }
    ln_fuse<<<NB * T, 32, 0, stream>>>(X, nullptr, nullptr, ng[i], nbb[i], X, T, D);
    head_dot<<<NB, 32, 0, stream>>>(X, hw[i], hb[i], OUTS, T * D, i);
    if (i < 2) {
      // merge: (56,T,D) -> (56,T/2,2D) @ W^T -> (56,T/2,D/2)
      launch_gemm(stream, X, 2 * D, 0, mw[i], mb[i], nullptr, Xalt, D / 2, 0,
                  NB * (T / 2), 2 * D, D / 2, 0);
      float* tp = X; X = Xalt; Xalt = tp;
      T >>= 1; D >>= 1;
    }
  }
  final_proj<<<1, 64, 0, stream>>>(OUTS, fw, fb, (float*)d_out);
}